// ETL_50337016709705
// MI455X (gfx1250) — compile-verified
//
#include <hip/hip_runtime.h>

typedef __attribute__((ext_vector_type(2))) float v2f;
typedef __attribute__((ext_vector_type(8))) float v8f;

#define BTOT    500000
#define EMB_DIM 64
#define HID     50
#define KPAD    196      // 193 padded to multiple of 4
#define KSTEPS  48       // embedding K-steps (t handled as step 48)
#define KPAIRS  98       // KPAD/2
#define PITCH   80       // float2 pitch per K-pair row: 160 words, mod 64 == 32 -> conflict-free
#define WAVES_PER_BLOCK 10
#define TPB     (WAVES_PER_BLOCK * 32)

__global__ __launch_bounds__(TPB) void fused_emb_mlp_wmma(
    const int*   __restrict__ batch_ind,  // [B,3]
    const float* __restrict__ batch_t,    // [B]
    const float* __restrict__ U,          // [300000,64]
    const float* __restrict__ W1,         // [193,50] row-major
    const float* __restrict__ b1,         // [50]
    const float* __restrict__ W2,         // [50]
    const float* __restrict__ b2,         // [1]
    float*       __restrict__ out)        // [B]
{
    // LDS: W1 staged as K-pairs: W1s[kp*PITCH + n] = (W1[2kp][n], W1[2kp+1][n]), n padded to 64
    __shared__ v2f W1s[KPAIRS * PITCH];   // 62,720 B

    const int tid = threadIdx.x;
    for (int i = tid; i < KPAIRS * 64; i += TPB) {
        const int kp = i >> 6;
        const int n  = i & 63;
        const int k0 = 2 * kp, k1 = 2 * kp + 1;
        v2f w;
        w.x = (k0 < 193 && n < HID) ? W1[k0 * HID + n] : 0.f;
        w.y = (k1 < 193 && n < HID) ? W1[k1 * HID + n] : 0.f;
        W1s[kp * PITCH + n] = w;
    }
    __syncthreads();

    const int lane = tid & 31;
    const int wave = tid >> 5;
    const int tile = blockIdx.x * WAVES_PER_BLOCK + wave;   // 0..31249
    const int m    = lane & 15;   // row within tile (A layout) == n col (B layout)
    const int h    = lane >> 4;   // half-wave: K offset selector
    const int row  = tile * 16 + m;

    // Per-row gather bases (mode offsets 0 / 100000 / 200000 into concatenated U)
    const float* rp0 = U + (size_t)(batch_ind[row * 3 + 0]         ) * EMB_DIM + 2 * h;
    const float* rp1 = U + (size_t)(batch_ind[row * 3 + 1] + 100000) * EMB_DIM + 2 * h;
    const float* rp2 = U + (size_t)(batch_ind[row * 3 + 2] + 200000) * EMB_DIM + 2 * h;
    const float  tval = batch_t[row];

    v8f acc0 = {}, acc1 = {}, acc2 = {}, acc3 = {};

    // K-steps 0..47: embeddings. mode = k/16 is wave-uniform; unrolled with constant indices.
    #pragma unroll
    for (int mode = 0; mode < 3; ++mode) {
        const float* p = (mode == 0) ? rp0 : (mode == 1) ? rp1 : rp2;
        #pragma unroll
        for (int kk = 0; kk < 16; ++kk) {
            v2f a;
            a.x = p[4 * kk];            // X[m][4k + 2h]
            a.y = p[4 * kk + 1];        // X[m][4k + 2h + 1]
            const int k  = mode * 16 + kk;
            const v2f* brow = &W1s[(2 * k + h) * PITCH + m];
            const v2f bb0 = brow[0];
            const v2f bb1 = brow[16];
            const v2f bb2 = brow[32];
            const v2f bb3 = brow[48];
            acc0 = __builtin_amdgcn_wmma_f32_16x16x4_f32(false, a, false, bb0, (short)0, acc0, false, false);
            acc1 = __builtin_amdgcn_wmma_f32_16x16x4_f32(false, a, false, bb1, (short)0, acc1, false, false);
            acc2 = __builtin_amdgcn_wmma_f32_16x16x4_f32(false, a, false, bb2, (short)0, acc2, false, false);
            acc3 = __builtin_amdgcn_wmma_f32_16x16x4_f32(false, a, false, bb3, (short)0, acc3, false, false);
        }
    }

    // K-step 48: the t column (feature 192), features 193..195 are zero pad.
    {
        v2f a;
        a.x = (h == 0) ? tval : 0.f;    // K=192 holds t, K=194 is pad
        a.y = 0.f;                      // K=193 / K=195 pad
        const v2f* brow = &W1s[(96 + h) * PITCH + m];
        const v2f bb0 = brow[0];
        const v2f bb1 = brow[16];
        const v2f bb2 = brow[32];
        const v2f bb3 = brow[48];
        acc0 = __builtin_amdgcn_wmma_f32_16x16x4_f32(false, a, false, bb0, (short)0, acc0, false, false);
        acc1 = __builtin_amdgcn_wmma_f32_16x16x4_f32(false, a, false, bb1, (short)0, acc1, false, false);
        acc2 = __builtin_amdgcn_wmma_f32_16x16x4_f32(false, a, false, bb2, (short)0, acc2, false, false);
        acc3 = __builtin_amdgcn_wmma_f32_16x16x4_f32(false, a, false, bb3, (short)0, acc3, false, false);
    }

    // Epilogue: out[m] = sum_n tanh(z[m][n] + b1[n]) * W2[n] + b2
    // C layout: lane holds (m = 8h + r, n = lane&15) for VGPR r = 0..7, per n-tile.
    float bbv[4], wwv[4];
    #pragma unroll
    for (int t2 = 0; t2 < 4; ++t2) {
        const int ng = t2 * 16 + m;
        const bool v = (ng < HID);
        bbv[t2] = v ? b1[ng] : 0.f;
        wwv[t2] = v ? W2[ng] : 0.f;
    }
    const float b2v = b2[0];

    float tot[8];
    #pragma unroll
    for (int r = 0; r < 8; ++r) {
        float s = tanhf(acc0[r] + bbv[0]) * wwv[0]
                + tanhf(acc1[r] + bbv[1]) * wwv[1]
                + tanhf(acc2[r] + bbv[2]) * wwv[2]
                + tanhf(acc3[r] + bbv[3]) * wwv[3];
        // reduce over the 16 lanes of this half-wave (N dimension)
        s += __shfl_xor(s, 1);
        s += __shfl_xor(s, 2);
        s += __shfl_xor(s, 4);
        s += __shfl_xor(s, 8);
        tot[r] = s;
    }

    if (m == 0) {
        float* o = out + tile * 16 + 8 * h;
        #pragma unroll
        for (int r = 0; r < 8; ++r) o[r] = tot[r] + b2v;
    }
}

extern "C" void kernel_launch(void* const* d_in, const int* in_sizes, int n_in,
                              void* d_out, int out_size, void* d_ws, size_t ws_size,
                              hipStream_t stream) {
    (void)in_sizes; (void)n_in; (void)out_size; (void)d_ws; (void)ws_size;
    const int*   batch_ind = (const int*)  d_in[0];
    const float* batch_t   = (const float*)d_in[1];
    const float* U         = (const float*)d_in[2];
    const float* W1        = (const float*)d_in[3];
    const float* b1        = (const float*)d_in[4];
    const float* W2        = (const float*)d_in[5];
    const float* b2        = (const float*)d_in[6];
    float* out = (float*)d_out;

    dim3 grid(BTOT / (16 * WAVES_PER_BLOCK));   // 3125
    dim3 block(TPB);                            // 320 = 10 waves
    hipLaunchKernelGGL(fused_emb_mlp_wmma, grid, block, 0, stream,
                       batch_ind, batch_t, U, W1, b1, W2, b2, out);
}